// MultiCamNoiseCropV3Prompter_45552423141950
// MI455X (gfx1250) — compile-verified
//
#include <hip/hip_runtime.h>

// Problem constants (match the reference)
#define P_   30
#define S_   224
#define CROP_ 164
#define B_   16
#define T_   16
#define W4_  (S_ / 4)   // 56 float4 per row

typedef float v4f __attribute__((ext_vector_type(4)));

// PyTorch/JAX bilinear (align_corners=False) 1D index/weight, float32-exact
// vs the reference: scale = in/out (correctly-rounded div), mul/add kept
// un-fused via explicit _rn intrinsics to avoid fma-contraction drift.
__device__ __forceinline__ void interp1d(float pos, int in_size, float out_size_f,
                                         int& i0, int& i1, float& w) {
    float scale = __fdiv_rn((float)in_size, out_size_f);
    float f = fmaxf(__fadd_rn(__fmul_rn(__fadd_rn(pos, 0.5f), scale), -0.5f), 0.0f);
    float fi = floorf(f);
    i0 = (int)fi;
    w  = __fsub_rn(f, fi);
    int ip = i0 + 1;
    int lim = in_size - 1;
    i1 = ip < lim ? ip : lim;
}

__device__ __forceinline__ float blend(float a, float b, float w) {
    // a*(1-w) + b*w with explicit round-to-nearest ops (no contraction)
    return __fadd_rn(__fmul_rn(a, __fsub_rn(1.0f, w)), __fmul_rn(b, w));
}

// ---------------------------------------------------------------------------
// Pass 1: build prompt[B][3][S][S] into workspace.
// grid = (S, B); block = S threads (224 = 7 waves). Thread j = column.
// Vertical region decided per-row (uniform per block); horizontal interp only
// needed on middle rows (vertical weight there is exactly 0 in the reference,
// so the row equals the horizontal interpolation verbatim).
// ---------------------------------------------------------------------------
__global__ void build_prompt_kernel(const float* __restrict__ pad_up,     // [3,3,2P,S]
                                    const float* __restrict__ pad_down,   // [3,3,P,S]
                                    const float* __restrict__ pad_left,   // [3,3,CROP,2P]
                                    const float* __restrict__ pad_right,  // [3,3,CROP,P]
                                    const int*   __restrict__ cam_views,  // [B]
                                    const int*   __restrict__ offs_r,     // [B]
                                    const int*   __restrict__ offs_d,     // [B]
                                    float*       __restrict__ prompt) {   // [B,3,S,S]
    const int i = blockIdx.x;        // output row
    const int b = blockIdx.y;        // sample
    const int j = threadIdx.x;       // output column

    const int cam   = cam_views[b];
    const int orr   = offs_r[b];
    const int od    = offs_d[b];
    const int off_l = 2 * P_ - orr;
    const int off_u = 2 * P_ - od;

    float out[3];

    if (i < off_u) {
        // top region: vertical interp within pad_up (rows 0..2P-1)
        int r0, r1; float w;
        interp1d((float)i, 2 * P_, (float)off_u, r0, r1, w);
        const float* src = pad_up + (size_t)cam * (3 * 2 * P_ * S_);
        #pragma unroll
        for (int c = 0; c < 3; ++c) {
            const float* pc = src + (size_t)c * (2 * P_ * S_);
            out[c] = blend(pc[r0 * S_ + j], pc[r1 * S_ + j], w);
        }
    } else if (i >= S_ - od) {
        // bottom region: vertical interp within pad_down (rows 0..P-1)
        int r0, r1; float w;
        interp1d((float)(i - (S_ - od)), P_, (float)od, r0, r1, w);
        const float* src = pad_down + (size_t)cam * (3 * P_ * S_);
        #pragma unroll
        for (int c = 0; c < 3; ++c) {
            const float* pc = src + (size_t)c * (P_ * S_);
            out[c] = blend(pc[r0 * S_ + j], pc[r1 * S_ + j], w);
        }
    } else {
        // middle region: horizontal interp of row y = i - off_u
        const int y = i - off_u;
        if (j < off_l) {
            int i0, i1; float w;
            interp1d((float)j, 2 * P_, (float)off_l, i0, i1, w);
            const float* src = pad_left + (size_t)cam * (3 * CROP_ * 2 * P_);
            #pragma unroll
            for (int c = 0; c < 3; ++c) {
                const float* pc = src + (size_t)c * (CROP_ * 2 * P_) + (size_t)y * (2 * P_);
                out[c] = blend(pc[i0], pc[i1], w);
            }
        } else if (j >= S_ - orr) {
            int i0, i1; float w;
            interp1d((float)(j - (S_ - orr)), P_, (float)orr, i0, i1, w);
            const float* src = pad_right + (size_t)cam * (3 * CROP_ * P_);
            #pragma unroll
            for (int c = 0; c < 3; ++c) {
                const float* pc = src + (size_t)c * (CROP_ * P_) + (size_t)y * P_;
                out[c] = blend(pc[i0], pc[i1], w);
            }
        } else {
            out[0] = out[1] = out[2] = 0.0f;
        }
    }

    #pragma unroll
    for (int c = 0; c < 3; ++c)
        prompt[(((size_t)b * 3 + c) * S_ + i) * S_ + j] = out[c];
}

// ---------------------------------------------------------------------------
// Pass 2: out = x + prompt (broadcast over T frames). Bandwidth pass.
// Each thread owns one float4 column-group of one (b, c, row) and loops over
// the 16 frames with the prompt float4 in registers. x/out use non-temporal
// 128-bit accesses (streamed once); next-t line prefetched (200KB stride).
// ---------------------------------------------------------------------------
__global__ void __launch_bounds__(256)
broadcast_add_kernel(const v4f* __restrict__ x,       // [B*3, T, S, W4]
                     const v4f* __restrict__ prompt,  // [B*3, S, W4]
                     v4f*       __restrict__ out) {
    const int id = blockIdx.x * blockDim.x + threadIdx.x;
    constexpr int total4 = B_ * 3 * S_ * W4_;          // 602112
    if (id >= total4) return;

    const int jj  = id % W4_;          // float4 column
    const int rem = id / W4_;
    const int i   = rem % S_;          // row
    const int bc  = rem / S_;          // (b*3 + c) plane, 0..47

    const v4f p = prompt[((size_t)bc * S_ + i) * W4_ + jj];

    const size_t stride = (size_t)S_ * W4_;                       // 12544 float4 per frame
    size_t idx = ((size_t)bc * T_) * stride + (size_t)i * W4_ + jj;

    #pragma unroll
    for (int t = 0; t < T_; ++t) {
        if (t + 1 < T_)
            __builtin_prefetch(&x[idx + stride], 0, 0);           // gfx1250 global_prefetch_b8
        v4f v = __builtin_nontemporal_load(&x[idx]);              // b128, th:NT
        v4f r = v + p;
        __builtin_nontemporal_store(r, &out[idx]);                // b128, th:NT
        idx += stride;
    }
}

// ---------------------------------------------------------------------------
extern "C" void kernel_launch(void* const* d_in, const int* in_sizes, int n_in,
                              void* d_out, int out_size, void* d_ws, size_t ws_size,
                              hipStream_t stream) {
    const float* x        = (const float*)d_in[0];
    const float* pad_up   = (const float*)d_in[1];
    const float* pad_down = (const float*)d_in[2];
    const float* pad_left = (const float*)d_in[3];
    const float* pad_right= (const float*)d_in[4];
    const int*   cam      = (const int*)d_in[5];
    const int*   offs_r   = (const int*)d_in[6];
    const int*   offs_d   = (const int*)d_in[7];

    float* prompt = (float*)d_ws;   // B*3*S*S floats = 9.63 MB, L2-resident

    dim3 g1(S_, B_);
    build_prompt_kernel<<<g1, S_, 0, stream>>>(pad_up, pad_down, pad_left, pad_right,
                                               cam, offs_r, offs_d, prompt);

    constexpr int total4 = B_ * 3 * S_ * W4_;          // 602112
    constexpr int block  = 256;
    constexpr int grid   = (total4 + block - 1) / block; // 2352
    broadcast_add_kernel<<<grid, block, 0, stream>>>((const v4f*)x, (const v4f*)prompt,
                                                     (v4f*)d_out);
}